// MLBMatchupPredictorTemporal_89455578841341
// MI455X (gfx1250) — compile-verified
//
#include <hip/hip_runtime.h>
#include <math.h>

typedef __attribute__((ext_vector_type(2))) float v2f;
typedef __attribute__((ext_vector_type(8))) float v8f;

constexpr int NN  = 20000;    // nodes per side (NP == NB)
constexpr int EED = 1000000;  // edges per direction
constexpr int EDT = 32;       // edge feature dim after temporal concat
constexpr int NC  = 5;
constexpr int CH  = 131072;   // edge chunk: 131072*64*4B = 32MB, L2-resident

// ---------- float <-> order-preserving unsigned (for atomic max) ----------
__device__ inline unsigned f2ord(float f) {
  unsigned u = __float_as_uint(f);
  return (u >> 31) ? ~u : (u | 0x80000000u);
}
__device__ inline float ord2f(unsigned u) {
  return (u >> 31) ? __uint_as_float(u & 0x7fffffffu) : __uint_as_float(~u);
}

// ---------- generic f32 WMMA GEMM: C[M,N] = A[M,K] @ B[K,N] (+bias) ----------
// Row-major everywhere. M%16==0, K%4==0, N%16==0. One wave per 16x16 C tile.
__global__ void gemm16(const float* __restrict__ A, const float* __restrict__ B,
                       const float* __restrict__ bias, float* __restrict__ C,
                       int M, int K, int N) {
  int wave   = blockIdx.x * (blockDim.x >> 5) + (threadIdx.x >> 5);
  int tilesN = N >> 4;
  int total  = (M >> 4) * tilesN;
  if (wave >= total) return;
  int tm   = wave / tilesN;
  int tn   = wave - tm * tilesN;
  int lane = threadIdx.x & 31;
  int half = lane >> 4;   // which K-pair of the 16x4 A fragment / 4x16 B fragment
  int l16  = lane & 15;
  int col  = tn * 16 + l16;

  const float* arow = A + (size_t)(tm * 16 + l16) * K + half * 2; // A[row, k0+2h : k0+2h+1]
  const float* bcol = B + (size_t)(half * 2) * N + col;           // B[k0+2h, col]
  v8f acc = {0.f, 0.f, 0.f, 0.f, 0.f, 0.f, 0.f, 0.f};
  for (int k0 = 0; k0 < K; k0 += 4) {
    v2f a = *(const v2f*)(arow + k0);
    v2f b;
    b.x = bcol[(size_t)k0 * N];
    b.y = bcol[(size_t)(k0 + 1) * N];
    acc = __builtin_amdgcn_wmma_f32_16x16x4_f32(false, a, false, b, (short)0, acc,
                                                false, false);
  }
  float bv = bias ? bias[col] : 0.0f;
  // D layout: acc[r] = C[tm*16 + r + 8*half][col]
  float* crow = C + (size_t)(tm * 16 + half * 8) * N + col;
#pragma unroll
  for (int r = 0; r < 8; ++r) crow[(size_t)r * N] = acc[r] + bv;
}

// ---------- edge feature + temporal encoding concat: [E,16] -> [E,32] ----------
__global__ void build_eafull(const float* __restrict__ ea, const float* __restrict__ t,
                             const float* __restrict__ freqs, float* __restrict__ out,
                             int Etot) {
  int tid = blockIdx.x * blockDim.x + threadIdx.x;
  if (tid >= Etot * 32) return;
  int e = tid >> 5, c = tid & 31;
  float v;
  if (c < 16) {
    v = ea[(size_t)e * 16 + c];
  } else {
    float w = t[e] * freqs[c & 7];
    v = (c < 24) ? sinf(w) : cosf(w);
  }
  out[tid] = v;
}

// ---------- GAT pass A: per-edge attention logit + segment max ----------
// One wave per edge. Lane c covers channel c (head0) and 32+c (head1).
__global__ void gat_logits(const float* __restrict__ xl, const float* __restrict__ xr,
                           const float* __restrict__ eedge, const int* __restrict__ ei,
                           const float* __restrict__ att, float* __restrict__ logit,
                           unsigned* __restrict__ mMax, int s, int cnt, int Etot) {
  int wid = blockIdx.x * (blockDim.x >> 5) + (threadIdx.x >> 5);
  if (wid >= cnt) return;
  int c = threadIdx.x & 31;
  int g = s + wid;
  int src = ei[g], dst = ei[Etot + g];
  const float* xls = xl + (size_t)src * 64;
  const float* xrd = xr + (size_t)dst * 64;
  const float* ee  = eedge + (size_t)wid * 64;
  float v0 = xls[c]      + xrd[c]      + ee[c];
  float v1 = xls[32 + c] + xrd[32 + c] + ee[32 + c];
  v0 = (v0 > 0.0f) ? v0 : 0.2f * v0;    // leaky_relu(0.2)
  v1 = (v1 > 0.0f) ? v1 : 0.2f * v1;
  v0 *= att[c];
  v1 *= att[32 + c];
  for (int off = 16; off; off >>= 1) {
    v0 += __shfl_xor(v0, off);
    v1 += __shfl_xor(v1, off);
  }
  if (c == 0) {
    logit[(size_t)g * 2 + 0] = v0;
    logit[(size_t)g * 2 + 1] = v1;
    atomicMax(mMax + (size_t)dst * 2 + 0, f2ord(v0));
    atomicMax(mMax + (size_t)dst * 2 + 1, f2ord(v1));
  }
}

// ---------- GAT pass B: ex = exp(logit - max), segment-sum denominator ----------
__global__ void gat_softmax(const int* __restrict__ ei, const unsigned* __restrict__ mMax,
                            float* __restrict__ exb, float* __restrict__ den, int Etot) {
  int tid = blockIdx.x * blockDim.x + threadIdx.x;
  if (tid >= Etot * 2) return;
  int e = tid >> 1, h = tid & 1;
  int dst = ei[Etot + e];
  float m  = ord2f(mMax[(size_t)dst * 2 + h]);
  float ex = expf(exb[(size_t)e * 2 + h] - m);
  exb[(size_t)e * 2 + h] = ex;
  atomicAdd(den + (size_t)dst * 2 + h, ex);
}

// ---------- GAT pass C: weighted scatter  acc[dst] += xl[src] * alpha ----------
__global__ void gat_scatter(const int* __restrict__ ei, const float* __restrict__ xl,
                            const float* __restrict__ exb, const float* __restrict__ den,
                            float* __restrict__ acc, int Etot) {
  int tid = blockIdx.x * blockDim.x + threadIdx.x;
  if (tid >= Etot * 64) return;
  int e = tid >> 6, c = tid & 63, h = c >> 5;
  int src = ei[e], dst = ei[Etot + e];
  float alpha = exb[(size_t)e * 2 + h] / (den[(size_t)dst * 2 + h] + 1e-16f);
  atomicAdd(acc + (size_t)dst * 64 + c, xl[(size_t)src * 64 + c] * alpha);
}

// ---------- GAT finalize: (+bias, relu) with optional head-mean ----------
__global__ void gat_finalize(const float* __restrict__ acc, const float* __restrict__ bias,
                             float* __restrict__ xout, int n, int concat) {
  int tid = blockIdx.x * blockDim.x + threadIdx.x;
  if (concat) {
    if (tid >= n * 64) return;
    int node = tid >> 6, c = tid & 63;
    xout[tid] = fmaxf(acc[(size_t)node * 64 + c] + bias[c], 0.0f);
  } else {
    if (tid >= n * 32) return;
    int node = tid >> 5, c = tid & 31;
    float v = 0.5f * (acc[(size_t)node * 64 + c] + acc[(size_t)node * 64 + 32 + c]) + bias[c];
    xout[tid] = fmaxf(v, 0.0f);
  }
}

// ---------- classifier head: one wave per edge, lane = channel ----------
__global__ void classifier_k(const int* __restrict__ ei, const float* __restrict__ xp2,
                             const float* __restrict__ xb2, const float* __restrict__ eaF,
                             const float* __restrict__ Wq, const float* __restrict__ bq,
                             const float* __restrict__ Wk, const float* __restrict__ bk,
                             const float* __restrict__ We, const float* __restrict__ be,
                             const float* __restrict__ W1, const float* __restrict__ b1,
                             const float* __restrict__ W2, const float* __restrict__ b2,
                             float* __restrict__ out, int Etot) {
  int e = blockIdx.x * (blockDim.x >> 5) + (threadIdx.x >> 5);
  if (e >= Etot) return;
  int c = threadIdx.x & 31;
  int src = ei[e], dst = ei[Etot + e];
  float sx = xp2[(size_t)src * 32 + c];
  float dx = xb2[(size_t)dst * 32 + c];
  float ev = eaF[(size_t)e * 32 + c];
  float q = bq[c], k = bk[c], ee = be[c];
#pragma unroll 8
  for (int j = 0; j < 32; ++j) {
    float sxj = __shfl(sx, j);
    float dxj = __shfl(dx, j);
    float evj = __shfl(ev, j);
    q  += sxj * Wq[j * 32 + c];
    k  += dxj * Wk[j * 32 + c];
    ee += evj * We[j * 32 + c];
  }
  float s = q * k;
  for (int off = 16; off; off >>= 1) s += __shfl_xor(s, off);
  float a = 1.0f / (1.0f + expf(-s));
  float comb = a * k + (1.0f - a) * q + ee;
  float h1 = b1[c];
#pragma unroll 8
  for (int j = 0; j < 32; ++j) h1 += __shfl(comb, j) * W1[j * 32 + c];
  h1 = fmaxf(h1, 0.0f);
  float o5 = 0.0f;
#pragma unroll 8
  for (int j = 0; j < 32; ++j) {
    float hj = __shfl(h1, j);
    if (c < NC) o5 += hj * W2[j * NC + c];
  }
  if (c < NC) out[(size_t)e * NC + c] = o5 + b2[c];
}

extern "C" void kernel_launch(void* const* d_in, const int* in_sizes, int n_in,
                              void* d_out, int out_size, void* d_ws, size_t ws_size,
                              hipStream_t stream) {
  (void)in_sizes; (void)n_in; (void)out_size; (void)ws_size;
  const float* x_p   = (const float*)d_in[0];
  const float* x_b   = (const float*)d_in[1];
  const int*   eifb  = (const int*)d_in[2];
  const int*   eibf  = (const int*)d_in[3];
  const float* eafb  = (const float*)d_in[4];
  const float* eabf  = (const float*)d_in[5];
  const float* tfb   = (const float*)d_in[6];
  const float* tbf   = (const float*)d_in[7];
  const float* freqs = (const float*)d_in[8];

  struct GatP { const float *Wl, *bl, *Wr, *br, *We, *att, *bias; };
  auto gp = [&](int b) -> GatP {
    return GatP{(const float*)d_in[b],     (const float*)d_in[b + 1],
                (const float*)d_in[b + 2], (const float*)d_in[b + 3],
                (const float*)d_in[b + 4], (const float*)d_in[b + 5],
                (const float*)d_in[b + 6]};
  };
  GatP g1fb = gp(9), g1bf = gp(16), g2fb = gp(23), g2bf = gp(30);
  const float* Wq  = (const float*)d_in[37];
  const float* bq  = (const float*)d_in[38];
  const float* Wk  = (const float*)d_in[39];
  const float* bk  = (const float*)d_in[40];
  const float* Wec = (const float*)d_in[41];
  const float* bec = (const float*)d_in[42];
  const float* W1c = (const float*)d_in[43];
  const float* b1c = (const float*)d_in[44];
  const float* W2c = (const float*)d_in[45];
  const float* b2c = (const float*)d_in[46];

  // -------- workspace layout (floats) --------
  float* ws = (float*)d_ws;
  size_t o = 0;
  float* eaFfb = ws + o; o += (size_t)EED * EDT;   // [E,32] fb edge feats (kept)
  float* eaFbf = ws + o; o += (size_t)EED * EDT;   // [E,32] bf edge feats (kept)
  float* eedge = ws + o; o += (size_t)CH * 64;     // chunk buffer for ea@We (L2)
  float* exbuf = ws + o; o += (size_t)EED * 2;     // logits, then exp values
  float* xlb   = ws + o; o += (size_t)NN * 64;
  float* xrb   = ws + o; o += (size_t)NN * 64;
  float* accb  = ws + o; o += (size_t)NN * 64;
  float* xb    = ws + o; o += (size_t)NN * 64;
  float* xp    = ws + o; o += (size_t)NN * 64;
  float* xb2   = ws + o; o += (size_t)NN * 32;
  float* xp2   = ws + o; o += (size_t)NN * 32;
  float* denb  = ws + o; o += (size_t)NN * 2;
  unsigned* mb = (unsigned*)(ws + o); o += (size_t)NN * 2;

  // -------- 1) temporal-encoded edge features --------
  int ethreads = EED * 32;
  build_eafull<<<(ethreads + 255) / 256, 256, 0, stream>>>(eafb, tfb, freqs, eaFfb, EED);
  build_eafull<<<(ethreads + 255) / 256, 256, 0, stream>>>(eabf, tbf, freqs, eaFbf, EED);

  // -------- 2) one GATv2 pass --------
  auto gat = [&](const float* xs, const float* xd, const int* ei, const float* eaF,
                 const GatP& P, int concat, float* xout) {
    int tiles = (NN / 16) * (64 / 16);  // 5000 waves
    gemm16<<<(tiles + 3) / 4, 128, 0, stream>>>(xs, P.Wl, P.bl, xlb, NN, 64, 64);
    gemm16<<<(tiles + 3) / 4, 128, 0, stream>>>(xd, P.Wr, P.br, xrb, NN, 64, 64);
    hipMemsetAsync(mb,   0, (size_t)NN * 2 * sizeof(unsigned), stream);
    hipMemsetAsync(denb, 0, (size_t)NN * 2 * sizeof(float), stream);
    hipMemsetAsync(accb, 0, (size_t)NN * 64 * sizeof(float), stream);
    for (int s = 0; s < EED; s += CH) {
      int cnt = (EED - s < CH) ? (EED - s) : CH;   // always a multiple of 16
      int t2  = (cnt / 16) * 4;                    // 16x16 tiles of [cnt,64]
      gemm16<<<(t2 + 3) / 4, 128, 0, stream>>>(eaF + (size_t)s * EDT, P.We, nullptr,
                                               eedge, cnt, EDT, 64);
      gat_logits<<<(cnt + 7) / 8, 256, 0, stream>>>(xlb, xrb, eedge, ei, P.att,
                                                    exbuf, mb, s, cnt, EED);
    }
    gat_softmax<<<(EED * 2 + 255) / 256, 256, 0, stream>>>(ei, mb, exbuf, denb, EED);
    gat_scatter<<<(EED * 64 + 255) / 256, 256, 0, stream>>>(ei, xlb, exbuf, denb, accb, EED);
    int nout = concat ? NN * 64 : NN * 32;
    gat_finalize<<<(nout + 255) / 256, 256, 0, stream>>>(accb, P.bias, xout, NN, concat);
  };

  gat(x_p, x_b, eifb, eaFfb, g1fb, 1, xb);   // layer 1 fb -> xb  [NN,64]
  gat(x_b, x_p, eibf, eaFbf, g1bf, 1, xp);   // layer 1 bf -> xp  [NN,64]
  gat(xp,  xb,  eifb, eaFfb, g2fb, 0, xb2);  // layer 2 fb -> xb2 [NN,32]
  gat(xb,  xp,  eibf, eaFbf, g2bf, 0, xp2);  // layer 2 bf -> xp2 [NN,32]

  // -------- 3) classifier head over fb edges --------
  classifier_k<<<(EED + 7) / 8, 256, 0, stream>>>(eifb, xp2, xb2, eaFfb,
                                                  Wq, bq, Wk, bk, Wec, bec,
                                                  W1c, b1c, W2c, b2c,
                                                  (float*)d_out, EED);
}